// MeanShiftTorch_68152541053157
// MI455X (gfx1250) — compile-verified
//
#include <hip/hip_runtime.h>

typedef float v2f __attribute__((ext_vector_type(2)));
typedef float v8f __attribute__((ext_vector_type(8)));

#define NPTS   2048
#define NTILE  (NPTS / 16)        // 128 column/row tiles
#define BLOCK  1024               // 32 wave32s on one WGP
#define NWAVE  (BLOCK / 32)
#define TPW    (NTILE / NWAVE)    // 4 row-tiles per wave

#define BW        0.05f
#define B2        (BW * BW)                 // 0.0025
#define THRESH2   2.5e-9f                   // (BW*0.001)^2
// -0.5/BW^2 * log2(e) : gaussian exponent prefolded for native v_exp_f32 (exp2)
#define NEXP2C    (-288.53900817779268f)

__device__ __forceinline__ v8f dist_tile_wmma(v2f a, v2f b) {
    v8f c = {};
    // (neg_a, A, neg_b, B, c_mod, C, reuse_a, reuse_b)
    c = __builtin_amdgcn_wmma_f32_16x16x4_f32(false, a, false, b, (short)0, c,
                                              false, false);
    return c;
}

__launch_bounds__(BLOCK, 1)
__global__ void meanshift_kernel(const float* __restrict__ feats,
                                 float* __restrict__ out) {
    // LDS state (single-WGP persistent kernel; everything lives here)
    __shared__ float Yaug[NPTS * 4];   // [y0, y1, y2, ||y||^2]  (B operand carries normY)
    __shared__ float Xt[2][NPTS * 4];  // [-2x0, -2x1, -2x2, ||x||^2] ping-pong
    __shared__ int   cntLds[NPTS];
    __shared__ int   dmaxBits;         // max delta^2 as monotonic int bits
    __shared__ int   selKey;           // (count<<12) | (2047-idx)

    const int tid  = threadIdx.x;
    const int lane = tid & 31;
    const int wave = tid >> 5;
    const int half = lane >> 4;   // which 16-lane half
    const int col  = lane & 15;   // N (column) / M (row) index within tile

    // ---- init: load 2048x3 feats into LDS ----
    for (int i = tid; i < NPTS; i += BLOCK) {
        float x0 = feats[i * 3 + 0];
        float x1 = feats[i * 3 + 1];
        float x2 = feats[i * 3 + 2];
        float nn = x0 * x0 + x1 * x1 + x2 * x2;
        Yaug[i * 4 + 0] = x0; Yaug[i * 4 + 1] = x1;
        Yaug[i * 4 + 2] = x2; Yaug[i * 4 + 3] = nn;
        Xt[0][i * 4 + 0] = -2.f * x0; Xt[0][i * 4 + 1] = -2.f * x1;
        Xt[0][i * 4 + 2] = -2.f * x2; Xt[0][i * 4 + 3] = nn;
    }
    if (tid == 0) { selKey = 0; }
    __syncthreads();

    // ---- mean-shift do-while loop (<= 11 steps, break when max||delta|| < thresh) ----
    int cur = 0;
    for (int it = 0; it < 11; ++it) {
        if (tid == 0) dmaxBits = 0;
        __syncthreads();

        const float* X  = Xt[cur];
        float*       Xn = Xt[cur ^ 1];
        float localDmax = 0.f;

        for (int rt = 0; rt < TPW; ++rt) {
            const int rowBase = (wave * TPW + rt) * 16;
            const int row     = rowBase + col;  // A: M = lane%16
            // A operand (16x4 f32: V0 holds K=2h, V1 holds K=2h+1):
            //   A row = [-2x0, -2x1, -2x2, 1]  (the +||x||^2 is applied per-row below)
            v2f a = *(const v2f*)&X[row * 4 + half * 2];
            if (half) a.y = 1.0f;               // K=3 element

            // per-row ||x||^2, hoisted out of the 128-iteration column loop
            float nX[8];
#pragma unroll
            for (int r = 0; r < 8; ++r)
                nX[r] = X[(rowBase + half * 8 + r) * 4 + 3];

            float acc[8][4];
#pragma unroll
            for (int r = 0; r < 8; ++r)
#pragma unroll
                for (int k = 0; k < 4; ++k) acc[r][k] = 0.f;

            for (int ct = 0; ct < NTILE; ++ct) {
                const int j = ct * 16 + col;    // B/C: N = lane%16
                float4 y = *(const float4*)&Yaug[j * 4];   // y0,y1,y2,||y||^2
                v2f b;
                b.x = half ? y.z : y.x;         // B: V0 = K(2h), V1 = K(2h+1)
                b.y = half ? y.w : y.y;         // K=3 element carries ||y||^2
                v8f c = dist_tile_wmma(a, b);   // c[r] = -2 x.y + ||y||^2
#pragma unroll
                for (int r = 0; r < 8; ++r) {
                    float d2 = c[r] + nX[r];                       // ||x - y||^2
                    float w  = __builtin_amdgcn_exp2f(NEXP2C * d2); // kernel (scale cancels)
                    acc[r][0] += w * y.x;
                    acc[r][1] += w * y.y;
                    acc[r][2] += w * y.z;
                    acc[r][3] += w;                                // denominator
                }
            }

            // butterfly-reduce across the 16 lanes of each half (columns)
#pragma unroll
            for (int m = 1; m <= 8; m <<= 1)
#pragma unroll
                for (int r = 0; r < 8; ++r)
#pragma unroll
                    for (int k = 0; k < 4; ++k)
                        acc[r][k] += __shfl_xor(acc[r][k], m, 32);

            // lanes col==r write row rowBase + 8*half + r
#pragma unroll
            for (int r = 0; r < 8; ++r) {
                if (col == r) {
                    const int orow = rowBase + half * 8 + r;
                    float inv = 1.f / acc[r][3];
                    float s0 = acc[r][0] * inv;
                    float s1 = acc[r][1] * inv;
                    float s2 = acc[r][2] * inv;
                    float o0 = -0.5f * X[orow * 4 + 0];
                    float o1 = -0.5f * X[orow * 4 + 1];
                    float o2 = -0.5f * X[orow * 4 + 2];
                    float e0 = s0 - o0, e1 = s1 - o1, e2 = s2 - o2;
                    float del2 = e0 * e0 + e1 * e1 + e2 * e2;
                    localDmax = fmaxf(localDmax, del2);
                    Xn[orow * 4 + 0] = -2.f * s0;
                    Xn[orow * 4 + 1] = -2.f * s1;
                    Xn[orow * 4 + 2] = -2.f * s2;
                    Xn[orow * 4 + 3] = s0 * s0 + s1 * s1 + s2 * s2;
                }
            }
        }

        atomicMax(&dmaxBits, __float_as_int(localDmax));  // delta^2 >= 0 -> int-monotonic
        __syncthreads();
        float dmax2 = __int_as_float(dmaxBits);
        __syncthreads();  // protect read against next-iteration reset
        cur ^= 1;         // Xn becomes current
        if (dmax2 < THRESH2) break;
    }

    // ---- neighbor counting on ORIGINAL feats (same WMMA distance tiles) ----
    for (int rt = 0; rt < TPW; ++rt) {
        const int rowBase = (wave * TPW + rt) * 16;
        const int row     = rowBase + col;
        float4 yr = *(const float4*)&Yaug[row * 4];
        v2f a;
        a.x = half ? (-2.f * yr.z) : (-2.f * yr.x);
        a.y = half ? 1.0f          : (-2.f * yr.y);

        float nR[8];
#pragma unroll
        for (int r = 0; r < 8; ++r)
            nR[r] = Yaug[(rowBase + half * 8 + r) * 4 + 3];

        int cnt[8];
#pragma unroll
        for (int r = 0; r < 8; ++r) cnt[r] = 0;

        for (int ct = 0; ct < NTILE; ++ct) {
            const int j = ct * 16 + col;
            float4 y = *(const float4*)&Yaug[j * 4];
            v2f b;
            b.x = half ? y.z : y.x;
            b.y = half ? y.w : y.y;
            v8f c = dist_tile_wmma(a, b);
#pragma unroll
            for (int r = 0; r < 8; ++r) cnt[r] += ((c[r] + nR[r]) < B2) ? 1 : 0;
        }
#pragma unroll
        for (int m = 1; m <= 8; m <<= 1)
#pragma unroll
            for (int r = 0; r < 8; ++r) cnt[r] += __shfl_xor(cnt[r], m, 32);
#pragma unroll
        for (int r = 0; r < 8; ++r)
            if (col == r) cntLds[rowBase + half * 8 + r] = cnt[r];
    }
    __syncthreads();

    // ---- argmax with first-max tie-break: key = (count<<12) | (2047-idx) ----
    for (int i = tid; i < NPTS; i += BLOCK) {
        int key = (cntLds[i] << 12) | (2047 - i);
        atomicMax(&selKey, key);
    }
    __syncthreads();
    const int maxIdx = 2047 - (selKey & 0xFFF);

    // ---- outputs: center coords (3 floats) + labels (2048 floats) ----
    const float cx = Yaug[maxIdx * 4 + 0];
    const float cy = Yaug[maxIdx * 4 + 1];
    const float cz = Yaug[maxIdx * 4 + 2];
    for (int j = tid; j < NPTS; j += BLOCK) {
        float dx = cx - Yaug[j * 4 + 0];
        float dy = cy - Yaug[j * 4 + 1];
        float dz = cz - Yaug[j * 4 + 2];
        float d2 = dx * dx + dy * dy + dz * dz;
        out[3 + j] = (d2 < B2) ? 1.0f : 0.0f;
    }
    if (tid < 3) out[tid] = -0.5f * Xt[cur][maxIdx * 4 + tid];
}

extern "C" void kernel_launch(void* const* d_in, const int* in_sizes, int n_in,
                              void* d_out, int out_size, void* d_ws, size_t ws_size,
                              hipStream_t stream) {
    (void)in_sizes; (void)n_in; (void)d_ws; (void)ws_size; (void)out_size;
    const float* feats = (const float*)d_in[0];
    float* out = (float*)d_out;
    // Single persistent workgroup: whole problem lives in one WGP's LDS,
    // iterations synchronized with workgroup barriers.
    meanshift_kernel<<<1, BLOCK, 0, stream>>>(feats, out);
}